// ParamMask_74620761801351
// MI455X (gfx1250) — compile-verified
//
#include <hip/hip_runtime.h>
#include <hip/hip_bf16.h>
#include <stdint.h>

// ---------------------------------------------------------------------------
// Top-k (50%) magnitude mask over 64M fp32 values.
//   u = float_bits & 0x7FFFFFFF is a monotone unsigned key for |x|.
//   Two-level radix select (14 high bits, then 17 low bits) finds exact
//   threshold T at ascending rank j = n/2; mask = (u >= T).
// Memory-bound: ~1 GB traffic -> ~44us at 23.3 TB/s.
// CDNA5 paths: 64KB LDS histogram + ds_add atomics, b128 streaming,
// async global->LDS staging (global_load_async_to_lds_b128 + s_wait_asynccnt).
// ---------------------------------------------------------------------------

#define HIST1_BITS 14
#define HIST1_BINS (1u << HIST1_BITS)          // 16384 bins, 64 KB LDS
#define HIST2_BITS 17
#define HIST2_BINS (1u << HIST2_BITS)          // 131072 bins, 512 KB global
#define LOW_MASK   (HIST2_BINS - 1u)

#if defined(__AMDGCN__)
#define USE_ASYNC_LDS 1
#else
#define USE_ASYNC_LDS 0
#endif

__device__ __forceinline__ uint32_t abs_key(float f) {
    return __float_as_uint(f) & 0x7FFFFFFFu;
}

#if USE_ASYNC_LDS
// LDS byte offset of a generic pointer that points into shared memory:
// addrspacecast generic->AS3, then ptrtoint (AS3 pointers are 32-bit).
__device__ __forceinline__ uint32_t lds_offset_u32(const void* p) {
    return (uint32_t)(uintptr_t)(const __attribute__((address_space(3))) void*)p;
}

__device__ __forceinline__ void async_load_b128_to_lds(const void* gaddr, uint32_t lds_addr) {
    // GV mode: vdst = LDS byte-address VGPR, v[pair] = 64-bit global address.
    asm volatile("global_load_async_to_lds_b128 %0, %1, off"
                 :
                 : "v"(lds_addr), "v"((uint64_t)(uintptr_t)gaddr)
                 : "memory");
}

__device__ __forceinline__ void wait_async0() {
#if __has_builtin(__builtin_amdgcn_s_wait_asynccnt)
    __builtin_amdgcn_s_wait_asynccnt(0);
#else
    asm volatile("s_wait_asynccnt 0" ::: "memory");
#endif
}
#endif

// ---- Pass 1: histogram of top 14 bits of |x| bit pattern (64 KB LDS/WG) ----
__global__ void hist1_kernel(const float4* __restrict__ in,
                             uint32_t* __restrict__ hist, size_t n4) {
    __shared__ uint32_t lh[HIST1_BINS];
    for (uint32_t i = threadIdx.x; i < HIST1_BINS; i += blockDim.x) lh[i] = 0u;
    __syncthreads();

    size_t stride = (size_t)gridDim.x * blockDim.x;
    for (size_t i = (size_t)blockIdx.x * blockDim.x + threadIdx.x; i < n4; i += stride) {
        float4 v = in[i];
        atomicAdd(&lh[abs_key(v.x) >> HIST2_BITS], 1u);
        atomicAdd(&lh[abs_key(v.y) >> HIST2_BITS], 1u);
        atomicAdd(&lh[abs_key(v.z) >> HIST2_BITS], 1u);
        atomicAdd(&lh[abs_key(v.w) >> HIST2_BITS], 1u);
    }
    __syncthreads();
    for (uint32_t i = threadIdx.x; i < HIST1_BINS; i += blockDim.x) {
        uint32_t c = lh[i];
        if (c) atomicAdd(&hist[i], c);
    }
}

// ---- Scan 1: locate the bin containing ascending rank j -------------------
// res[0] = selected high bin, res[1] = residual rank inside that bin
__global__ void scan1_kernel(const uint32_t* __restrict__ hist,
                             uint32_t* __restrict__ res, uint32_t j) {
    __shared__ uint32_t partial[256];
    const uint32_t CHUNK = HIST1_BINS / 256u;
    uint32_t s = 0;
    uint32_t base = threadIdx.x * CHUNK;
    for (uint32_t i = 0; i < CHUNK; ++i) s += hist[base + i];
    partial[threadIdx.x] = s;
    __syncthreads();
    if (threadIdx.x == 0) {
        uint64_t cum = 0;
        uint32_t c = 0;
        for (; c < 256u; ++c) {
            if (cum + partial[c] > (uint64_t)j) break;
            cum += partial[c];
        }
        uint32_t b = c * CHUNK;
        for (;; ++b) {
            uint32_t h = hist[b];
            if (cum + h > (uint64_t)j) break;
            cum += h;
        }
        res[0] = b;
        res[1] = (uint32_t)((uint64_t)j - cum);
    }
}

// ---- Pass 2: histogram low 17 bits of elements in the selected high bin ---
__global__ void hist2_kernel(const float4* __restrict__ in,
                             const uint32_t* __restrict__ res,
                             uint32_t* __restrict__ hist2, size_t n4) {
    uint32_t b1 = res[0];
    size_t stride = (size_t)gridDim.x * blockDim.x;
    for (size_t i = (size_t)blockIdx.x * blockDim.x + threadIdx.x; i < n4; i += stride) {
        float4 v = in[i];
        uint32_t u;
        u = abs_key(v.x); if ((u >> HIST2_BITS) == b1) atomicAdd(&hist2[u & LOW_MASK], 1u);
        u = abs_key(v.y); if ((u >> HIST2_BITS) == b1) atomicAdd(&hist2[u & LOW_MASK], 1u);
        u = abs_key(v.z); if ((u >> HIST2_BITS) == b1) atomicAdd(&hist2[u & LOW_MASK], 1u);
        u = abs_key(v.w); if ((u >> HIST2_BITS) == b1) atomicAdd(&hist2[u & LOW_MASK], 1u);
    }
}

// ---- Scan 2: exact 31-bit threshold ---------------------------------------
// res[2] = T
__global__ void scan2_kernel(const uint32_t* __restrict__ hist2,
                             uint32_t* __restrict__ res) {
    __shared__ uint32_t partial[256];
    const uint32_t CHUNK = HIST2_BINS / 256u;
    uint32_t rank = res[1];
    uint32_t b1   = res[0];
    uint32_t s = 0;
    uint32_t base = threadIdx.x * CHUNK;
    for (uint32_t i = 0; i < CHUNK; ++i) s += hist2[base + i];
    partial[threadIdx.x] = s;
    __syncthreads();
    if (threadIdx.x == 0) {
        uint64_t cum = 0;
        uint32_t c = 0;
        for (; c < 256u; ++c) {
            if (cum + partial[c] > (uint64_t)rank) break;
            cum += partial[c];
        }
        uint32_t b = c * CHUNK;
        for (;; ++b) {
            uint32_t h = hist2[b];
            if (cum + h > (uint64_t)rank) break;
            cum += h;
        }
        res[2] = (b1 << HIST2_BITS) | b;
    }
}

// ---- Pass 3: write mask = (|x| >= T), streamed as float4 ------------------
__global__ void mask_kernel(const float4* __restrict__ in,
                            float4* __restrict__ out,
                            const uint32_t* __restrict__ res, size_t n4) {
#if USE_ASYNC_LDS
    __shared__ float4 tile[256];
#endif
    uint32_t T = res[2];
    size_t stride = (size_t)gridDim.x * blockDim.x;
    for (size_t i = (size_t)blockIdx.x * blockDim.x + threadIdx.x; i < n4; i += stride) {
        float4 v;
#if USE_ASYNC_LDS
        // Stage this lane's 16B through LDS via the CDNA5 async-copy engine.
        async_load_b128_to_lds(in + i, lds_offset_u32(&tile[threadIdx.x]));
        wait_async0();
        asm volatile("" ::: "memory");
        v = tile[threadIdx.x];
#else
        v = in[i];
#endif
        float4 r;
        r.x = (abs_key(v.x) >= T) ? 1.0f : 0.0f;
        r.y = (abs_key(v.y) >= T) ? 1.0f : 0.0f;
        r.z = (abs_key(v.z) >= T) ? 1.0f : 0.0f;
        r.w = (abs_key(v.w) >= T) ? 1.0f : 0.0f;
        out[i] = r;
    }
}

// ---------------------------------------------------------------------------
extern "C" void kernel_launch(void* const* d_in, const int* in_sizes, int n_in,
                              void* d_out, int out_size, void* d_ws, size_t ws_size,
                              hipStream_t stream) {
    (void)n_in; (void)out_size; (void)ws_size;
    const float*  param = (const float*)d_in[0];
    const size_t  n     = (size_t)in_sizes[0];     // 4096*16384 = 67108864
    const size_t  n4    = n / 4;
    const uint32_t rank = (uint32_t)(n / 2);       // j = (1-0.5)*n

    uint32_t* res   = (uint32_t*)d_ws;                                   // [0..7]
    uint32_t* hist1 = (uint32_t*)((char*)d_ws + 1024);                   // 64 KB
    uint32_t* hist2 = (uint32_t*)((char*)d_ws + 1024 + HIST1_BINS * 4);  // 512 KB

    const size_t scratch_bytes = 1024 + (size_t)(HIST1_BINS + HIST2_BINS) * 4;
    (void)hipMemsetAsync(d_ws, 0, scratch_bytes, stream);

    const int threads = 256;
    const int blocks  = 4096;

    hist1_kernel<<<blocks, threads, 0, stream>>>((const float4*)param, hist1, n4);
    scan1_kernel<<<1, 256, 0, stream>>>(hist1, res, rank);
    hist2_kernel<<<blocks, threads, 0, stream>>>((const float4*)param, res, hist2, n4);
    scan2_kernel<<<1, 256, 0, stream>>>(hist2, res);
    mask_kernel <<<blocks, threads, 0, stream>>>((const float4*)param, (float4*)d_out, res, n4);
}